// MonotonicAttention_68461778698926
// MI455X (gfx1250) — compile-verified
//
#include <hip/hip_runtime.h>
#include <math.h>

// MI455X / gfx1250 monotonic attention forward.
// d_out = [context (B*H floats)] ++ [alpha (B*T floats)]
// ws usage: dec_proj (B*A floats = 8KB) + context partials (B*16*H floats = 512KB)

typedef __attribute__((ext_vector_type(2))) float v2f;
typedef __attribute__((ext_vector_type(4))) float v4f;
typedef __attribute__((ext_vector_type(8))) float v8f;

#define B_    16
#define T_    4096
#define H_    512
#define A_    128
#define DEC_  512
#define EPS_  1e-6f

// ---- Kernel 1 tiling ----
#define KC            32                 // K chunk staged in LDS
#define ROWS_PER_BLK  128                // 8 waves * 16 rows
#define LDSA_STRIDE   (KC + 4)           // 36: r*36 mod 64 distinct over r=0..15 -> conflict-free
#define LDSBP_STRIDE  (2 * A_ + 32)      // 288 ≡ 32 (mod 64): lane-halves hit disjoint bank ranges

#if __has_builtin(__builtin_amdgcn_tanhf)
  #define TANH_F32(x) __builtin_amdgcn_tanhf(x)
#else
  #define TANH_F32(x) tanhf(x)
#endif

#define TCHUNK  256
#define NCHUNK  (T_ / TCHUNK)            // 16 context partials per batch

// ---------------- K0: dec_proj[b][n] = dec[b] @ W_s[:,n] + b_s[n] ----------------
__global__ __launch_bounds__(A_)
void dec_proj_kernel(const float* __restrict__ dec, const float* __restrict__ W_s,
                     const float* __restrict__ b_s, float* __restrict__ dp) {
  const int b = blockIdx.x;
  const int n = threadIdx.x;
  float s = b_s[n];
  for (int h = 0; h < DEC_; ++h)
    s += dec[b * DEC_ + h] * W_s[h * A_ + n];
  dp[b * A_ + n] = s;
}

// ---------------- K1: fused enc@W_h (f32 WMMA) + tanh + dot(v) + mask + sigmoid -> p ----------------
__global__ __launch_bounds__(256)
void fused_proj_score(const float* __restrict__ enc,
                      const int*   __restrict__ enc_lens,
                      const float* __restrict__ W_h,
                      const float* __restrict__ dp,
                      const float* __restrict__ v,
                      float* __restrict__ p_out /* alpha region of d_out, [B][T] */) {
  __shared__ float lA[ROWS_PER_BLK * LDSA_STRIDE];   // 128*36*4  = 18432 B
  __shared__ float lB[(KC / 2) * LDSBP_STRIDE];      // 16*288*4  = 18432 B

  const int tid  = threadIdx.x;
  const int wave = tid >> 5;
  const int lane = tid & 31;
  const int l16  = lane & 15;
  const int hi   = lane >> 4;

  const int rowBlock = blockIdx.x * ROWS_PER_BLK;    // 16-row tiles never cross batch (T%16==0)
  const int rowBase  = rowBlock + wave * 16;

  v8f acc[8] = {};                                   // N = 8 tiles of 16 (all of A_=128)

  for (int k0 = 0; k0 < H_; k0 += KC) {
    __syncthreads();
    // stage A: 128 rows x KC floats of enc (coalesced float4)
    {
      const int r0 = tid >> 3;                       // 0..31
      const int c4 = (tid & 7) * 4;                  // 0..28
      #pragma unroll
      for (int rr = 0; rr < ROWS_PER_BLK; rr += 32) {
        v4f x = *(const v4f*)(enc + (size_t)(rowBlock + r0 + rr) * H_ + k0 + c4);
        *(v4f*)(lA + (r0 + rr) * LDSA_STRIDE + c4) = x;
      }
    }
    // stage B: KC rows x 128 cols of W_h, k-pair interleaved:
    //   lB[kpair*STRIDE + 2n + (k&1)] = W_h[k0 + 2*kpair + (k&1)][n]
    // -> per-lane B fragments become one contiguous 8-byte LDS load.
    {
      const int l   = tid & 63;                      // n pair index: n = 2l, 2l+1
      const int kp0 = tid >> 6;                      // 0..3
      #pragma unroll
      for (int kp = 0; kp < KC / 2; kp += 4) {
        const int k = k0 + 2 * (kp0 + kp);
        v2f w0 = *(const v2f*)(W_h + (size_t)k * A_ + 2 * l);        // W[k][2l], W[k][2l+1]
        v2f w1 = *(const v2f*)(W_h + (size_t)(k + 1) * A_ + 2 * l);  // W[k+1][2l], W[k+1][2l+1]
        v4f o; o.x = w0.x; o.y = w1.x; o.z = w0.y; o.w = w1.y;
        *(v4f*)(lB + (kp0 + kp) * LDSBP_STRIDE + 4 * l) = o;
      }
    }
    __syncthreads();

    #pragma unroll
    for (int s = 0; s < KC; s += 4) {
      // A frag: lane<16 -> (M=l16, K={s,s+1}); lane>=16 -> (M=l16, K={s+2,s+3})
      v2f afrag = *(const v2f*)(lA + (wave * 16 + l16) * LDSA_STRIDE + s + 2 * hi);
      // B frag: kpair = s/2 + hi; v0 = W[k][n], v1 = W[k+1][n], contiguous in LDS
      const float* bbase = lB + (s / 2 + hi) * LDSBP_STRIDE + 2 * l16;
      #pragma unroll
      for (int t = 0; t < 8; ++t) {
        v2f bfrag = *(const v2f*)(bbase + t * 32);
        acc[t] = __builtin_amdgcn_wmma_f32_16x16x4_f32(
            false, afrag, false, bfrag, (short)0, acc[t], false, false);
      }
    }
  }

  // Epilogue: e[m] = sum_n tanh(proj[m][n] + dp[b][n]) * v[n]; mask; sigmoid.
  const int bIdx = rowBase >> 12;                    // rowBase / T_
  const int len  = enc_lens[bIdx];
  float dpv[8], vv[8];
  #pragma unroll
  for (int t = 0; t < 8; ++t) {
    dpv[t] = dp[bIdx * A_ + t * 16 + l16];
    vv[t]  = v[t * 16 + l16];
  }
  #pragma unroll
  for (int r = 0; r < 8; ++r) {
    float part = 0.0f;
    #pragma unroll
    for (int t = 0; t < 8; ++t)
      part += TANH_F32(acc[t][r] + dpv[t]) * vv[t];
    // reduce across the 16-lane half (lanes 0-15 -> row r, lanes 16-31 -> row r+8)
    #pragma unroll
    for (int off = 1; off < 16; off <<= 1)
      part += __shfl_xor(part, off, 32);
    if (l16 == r) {
      const int m    = rowBase + r + 8 * hi;
      const int tpos = m & (T_ - 1);
      float pv = (tpos >= len) ? 0.0f : 1.0f / (1.0f + __expf(-part));
      p_out[m] = pv;                                 // sigmoid(masked e) == 0 exactly
    }
  }
}

// ---------------- K2: wave-parallel linear-recurrence scan; p -> alpha in place ----------------
__global__ __launch_bounds__(32)
void scan_kernel(const float* __restrict__ alpha_prev, float* pa /* p in, alpha out */) {
  const int b    = blockIdx.x;
  const int lane = threadIdx.x;
  const int CS   = T_ / 32;                          // 128 per lane
  const int cs   = lane * CS;
  const float* pb  = pa + b * T_;
  const float* apb = alpha_prev + b * T_;

  const float pboundary = (cs == 0) ? 0.0f : pb[cs - 1];

  // pass 1: fold chunk into affine (Ac, Bc): y_end = Ac*y_in + Bc
  float Ac = 1.0f, Bc = 0.0f, pprev = pboundary;
  for (int j = cs; j < cs + CS; ++j) {
    float pj = pb[j];
    float aj = (j == 0) ? 0.0f : pj * (1.0f - pprev) / (pprev + EPS_);
    float bj = pj * apb[j];
    Ac = aj * Ac;
    Bc = aj * Bc + bj;
    pprev = pj;
  }
  // inclusive shuffle scan of affine composition across lanes
  #pragma unroll
  for (int off = 1; off < 32; off <<= 1) {
    float As = __shfl_up(Ac, off, 32);
    float Bs = __shfl_up(Bc, off, 32);
    if (lane >= off) { Bc = Ac * Bs + Bc; Ac = Ac * As; }
  }
  float yin = __shfl_up(Bc, 1, 32);
  if (lane == 0) yin = 0.0f;                         // alpha_{-1} := 0 (a_0 = 0 makes alpha_0 = b_0)

  // pass 2: replay chunk, overwrite p with alpha (read-before-write per element)
  float y = yin;
  pprev = pboundary;
  float* ab = pa + b * T_;
  for (int j = cs; j < cs + CS; ++j) {
    float pj = pb[j];
    float aj = (j == 0) ? 0.0f : pj * (1.0f - pprev) / (pprev + EPS_);
    float bj = pj * apb[j];
    y = aj * y + bj;
    ab[j] = y;
    pprev = pj;
  }
}

// ---------------- K3: deterministic context partials over T chunks ----------------
__global__ __launch_bounds__(H_)
void context_partial(const float* __restrict__ enc, const float* __restrict__ alpha,
                     float* __restrict__ part) {
  const int chunk = blockIdx.x;                      // 0..NCHUNK-1
  const int b     = blockIdx.y;                      // 0..B-1
  const int h     = threadIdx.x;                     // 0..H-1
  const int t0    = chunk * TCHUNK;
  const float* e  = enc + ((size_t)b * T_ + t0) * H_ + h;
  const float* al = alpha + b * T_ + t0;
  float s = 0.0f;
  for (int t = 0; t < TCHUNK; ++t)
    s += al[t] * e[(size_t)t * H_];
  part[(b * NCHUNK + chunk) * H_ + h] = s;
}

// ---------------- K4: reduce partials -> context ----------------
__global__ __launch_bounds__(H_)
void context_reduce(const float* __restrict__ part, float* __restrict__ ctx) {
  const int b = blockIdx.x;
  const int h = threadIdx.x;
  float s = 0.0f;
  #pragma unroll
  for (int c = 0; c < NCHUNK; ++c)
    s += part[(b * NCHUNK + c) * H_ + h];
  ctx[b * H_ + h] = s;
}

extern "C" void kernel_launch(void* const* d_in, const int* in_sizes, int n_in,
                              void* d_out, int out_size, void* d_ws, size_t ws_size,
                              hipStream_t stream) {
  (void)in_sizes; (void)n_in; (void)out_size; (void)ws_size;
  const float* enc      = (const float*)d_in[0];
  const int*   enc_lens = (const int*)  d_in[1];
  const float* dec      = (const float*)d_in[2];
  const float* ap       = (const float*)d_in[3];
  const float* W_h      = (const float*)d_in[4];
  const float* W_s      = (const float*)d_in[5];
  const float* b_s      = (const float*)d_in[6];
  const float* v        = (const float*)d_in[7];

  float* ctx   = (float*)d_out;                      // [B][H]
  float* alpha = (float*)d_out + B_ * H_;            // [B][T] (holds p between K1 and K2)

  float* dp   = (float*)d_ws;                        // [B][A]
  float* part = dp + B_ * A_;                        // [B][NCHUNK][H]

  dec_proj_kernel<<<B_, A_, 0, stream>>>(dec, W_s, b_s, dp);
  fused_proj_score<<<(B_ * T_) / ROWS_PER_BLK, 256, 0, stream>>>(enc, enc_lens, W_h, dp, v, alpha);
  scan_kernel<<<B_, 32, 0, stream>>>(ap, alpha);
  dim3 g3(NCHUNK, B_);
  context_partial<<<g3, H_, 0, stream>>>(enc, alpha, part);
  context_reduce<<<B_, H_, 0, stream>>>(part, ctx);
}